// Layer_15625091023064
// MI455X (gfx1250) — compile-verified
//
#include <hip/hip_runtime.h>
#include <hip/hip_bf16.h>
#include <cstddef>

typedef __attribute__((ext_vector_type(2))) float v2f;
typedef __attribute__((ext_vector_type(8))) float v8f;

#define D 128
#define PAD_STRIDE 132     // 128 + 4 pad floats: 528B rows (16B aligned), conflict-free b64 LDS reads
#define MB 64              // M rows per block (4 sub-tiles of 16)
#define W_LDS_FLOATS (D * PAD_STRIDE)       // 16896
#define X_LDS_FLOATS (MB * PAD_STRIDE)      // 8448
#define SUB_STRIDE (16 * PAD_STRIDE)        // floats between M-subtiles in LDS

// Hardware float atomic max (GLOBAL_ATOMIC_MAX_NUM_F32, non-returning, device scope).
__device__ __forceinline__ void gmax_f32(float* addr, float v) {
    asm volatile("global_atomic_max_num_f32 %0, %1, off scope:SCOPE_DEV"
                 :: "v"(addr), "v"(v) : "memory");
}

// ---------------------------------------------------------------------------
// Kernel 1: fill agg with -inf (float4 stores)
// ---------------------------------------------------------------------------
__global__ __launch_bounds__(256) void init_agg(float4* __restrict__ agg, int n4) {
    int i = blockIdx.x * blockDim.x + threadIdx.x;
    if (i < n4) {
        const float ninf = -__builtin_inff();
        agg[i] = make_float4(ninf, ninf, ninf, ninf);
    }
}

// ---------------------------------------------------------------------------
// Kernel 2: symmetric scatter-max. One wave32 per edge; lane l owns dims
// [4l, 4l+4). Coalesced float4 row gathers (L2-resident), 8 atomics/lane.
// ---------------------------------------------------------------------------
__global__ __launch_bounds__(256) void scatter_max(const float* __restrict__ h,
                                                   const long long* __restrict__ ei,
                                                   float* __restrict__ agg,
                                                   int nedges) {
    int gid  = blockIdx.x * blockDim.x + threadIdx.x;
    int e    = gid >> 5;
    int lane = gid & 31;
    if (e >= nedges) return;

    long long u = ei[e];
    long long v = ei[(size_t)nedges + e];

    const float4 hu = *(const float4*)(h + (size_t)u * D + lane * 4);
    const float4 hv = *(const float4*)(h + (size_t)v * D + lane * 4);

    float* av = agg + (size_t)v * D + lane * 4;
    float* au = agg + (size_t)u * D + lane * 4;

    gmax_f32(av + 0, hu.x); gmax_f32(av + 1, hu.y);
    gmax_f32(av + 2, hu.z); gmax_f32(av + 3, hu.w);
    gmax_f32(au + 0, hv.x); gmax_f32(au + 1, hv.y);
    gmax_f32(au + 2, hv.z); gmax_f32(au + 3, hv.w);
}

__device__ __forceinline__ float decode_add(float hv, float gv, float ninf) {
    return hv + (gv > ninf ? gv : 0.0f);
}

// ---------------------------------------------------------------------------
// Kernel 3: out = (h + agg') @ W^T + b with V_WMMA_F32_16X16X4_F32.
// Block = 256 threads = 8 waves (one N-tile each), MB=64 M-rows per block.
// W and the decoded X tile are staged once per block in padded LDS; each
// wave runs 4 independent accumulator chains (4 M-subtiles), reusing each
// B fragment 4x and hiding WMMA accumulation latency.
// ---------------------------------------------------------------------------
__global__ __launch_bounds__(256) void fused_gemm(const float* __restrict__ h,
                                                  const float* __restrict__ agg,
                                                  const float* __restrict__ W,
                                                  const float* __restrict__ b,
                                                  float* __restrict__ out,
                                                  int nnodes) {
    extern __shared__ float lds[];
    float* lw = lds;                 // W: 128 x PAD_STRIDE
    float* lx = lds + W_LDS_FLOATS;  // X: 64 x PAD_STRIDE

    const int tid  = threadIdx.x;
    const int m0   = blockIdx.x * MB;
    const float ninf = -__builtin_inff();

    // --- Stage W (64KB): 2 threads per row, 16 float4 each, coalesced. ---
    {
        const int row  = tid >> 1;
        const int half = tid & 1;
        const float4* src = (const float4*)(W + row * D + half * 64);
        float4*       dst = (float4*)(lw + row * PAD_STRIDE + half * 64);
#pragma unroll
        for (int j = 0; j < 16; ++j) dst[j] = src[j];
    }

    // --- Stage decoded X = h + (agg > -inf ? agg : 0): 64 rows x 128. ---
    // 4 threads per row, 8 float4 pairs each; decode done ONCE per element.
    {
        const int row   = tid >> 2;                    // 0..63
        const int cbase = (tid & 3) * 32;              // 0,32,64,96
        const int rg    = min(m0 + row, nnodes - 1);   // clamp for partial tail
        const float4* hp = (const float4*)(h   + (size_t)rg * D + cbase);
        const float4* gp = (const float4*)(agg + (size_t)rg * D + cbase);
        float4*       xp = (float4*)(lx + row * PAD_STRIDE + cbase);
#pragma unroll
        for (int j = 0; j < 8; ++j) {
            float4 hv = hp[j];
            float4 gv = gp[j];
            float4 x;
            x.x = decode_add(hv.x, gv.x, ninf);
            x.y = decode_add(hv.y, gv.y, ninf);
            x.z = decode_add(hv.z, gv.z, ninf);
            x.w = decode_add(hv.w, gv.w, ninf);
            xp[j] = x;
        }
    }
    __syncthreads();

    const int wave  = tid >> 5;          // 0..7 -> N tile
    const int lane  = tid & 31;
    const int n0    = wave * 16;
    const int lrow  = lane & 15;
    const int khalf = lane >> 4;         // 0 or 1

    // Fragment base pointers (LDS). A frag for subtile s lives at +s*SUB_STRIDE.
    const float* axp = lx + lrow * PAD_STRIDE + 2 * khalf;
    const float* bwp = lw + (n0 + lrow) * PAD_STRIDE + 2 * khalf;

    v8f acc[4];
#pragma unroll
    for (int s = 0; s < 4; ++s) acc[s] = (v8f){};

#pragma unroll 4
    for (int k = 0; k < D; k += 4) {
        v2f bv = *(const v2f*)(bwp + k);          // B frag: reused by 4 chains
#pragma unroll
        for (int s = 0; s < 4; ++s) {
            v2f av = *(const v2f*)(axp + s * SUB_STRIDE + k);
            acc[s] = __builtin_amdgcn_wmma_f32_16x16x4_f32(
                         false, av, false, bv, (short)0, acc[s], false, false);
        }
    }

    // --- Epilogue: +bias, store. VGPR r of lane l -> (M = mt + r + 8*khalf,
    // N = n0 + (lane&15)). Wave-uniform guard; per-element only on tail tile.
    const float bias = b[n0 + lrow];
#pragma unroll
    for (int s = 0; s < 4; ++s) {
        const int mt = m0 + s * 16;
        const int mb = mt + khalf * 8;
        float* op = out + (size_t)mb * D + n0 + lrow;
        if (mt + 16 <= nnodes) {
#pragma unroll
            for (int r = 0; r < 8; ++r)
                op[(size_t)r * D] = acc[s][r] + bias;
        } else {
#pragma unroll
            for (int r = 0; r < 8; ++r)
                if (mb + r < nnodes)
                    op[(size_t)r * D] = acc[s][r] + bias;
        }
    }
}

// ---------------------------------------------------------------------------
extern "C" void kernel_launch(void* const* d_in, const int* in_sizes, int n_in,
                              void* d_out, int out_size, void* d_ws, size_t ws_size,
                              hipStream_t stream) {
    const float*     h  = (const float*)d_in[0];
    const long long* ei = (const long long*)d_in[1];   // int64 edge_index [2, E]
    const float*     W  = (const float*)d_in[2];
    const float*     b  = (const float*)d_in[3];
    float*           o  = (float*)d_out;
    float*           agg = (float*)d_ws;               // N * D floats (25.6 MB)

    const int N = in_sizes[0] / D;   // 50000
    const int E = in_sizes[1] / 2;   // 600000

    // 1) agg = -inf
    {
        int n4 = (N * D) / 4;
        int blocks = (n4 + 255) / 256;
        init_agg<<<blocks, 256, 0, stream>>>((float4*)agg, n4);
    }
    // 2) symmetric scatter-max (one wave per edge)
    {
        long long threads = (long long)E * 32;
        int blocks = (int)((threads + 255) / 256);
        scatter_max<<<blocks, 256, 0, stream>>>(h, ei, agg, E);
    }
    // 3) fused (h + agg) @ W^T + b with fp32 WMMA, LDS-staged operands
    {
        int blocks = (N + MB - 1) / MB;   // 782
        size_t lds_bytes = (size_t)(W_LDS_FLOATS + X_LDS_FLOATS) * sizeof(float); // 101376
        fused_gemm<<<blocks, 256, lds_bytes, stream>>>(h, agg, W, b, o, N);
    }
}